// CohereDecoderWithCache_84774064488703
// MI455X (gfx1250) — compile-verified
//
#include <hip/hip_runtime.h>
#include <hip/hip_bf16.h>

// Problem constants (must match reference)
#define LNUM   8
#define HNUM   8
#define DHEAD  128
#define MAXP   512
#define EMB    1024
#define DFFN   4096
#define ENCL   1024
#define BATCH  8

typedef float v2f __attribute__((ext_vector_type(2)));
typedef float v8f __attribute__((ext_vector_type(8)));

// workspace layout (float offsets)
#define WS_X    0
#define WS_Y    (16*1024)
#define WS_Q    (2*16*1024)
#define WS_K    (3*16*1024)
#define WS_V    (4*16*1024)
#define WS_ATTN (5*16*1024)
#define WS_H1   (6*16*1024)                 // 16 x 4096
#define WS_META (6*16*1024 + 16*4096)       // ints: widx[8], valid_max at [8]

// ---------------------------------------------------------------------------
// small helpers
// ---------------------------------------------------------------------------
__device__ __forceinline__ float block_reduce_sum256(float v, float* red) {
    int t = threadIdx.x;
    red[t] = v; __syncthreads();
    for (int s = 128; s > 0; s >>= 1) {
        if (t < s) red[t] += red[t + s];
        __syncthreads();
    }
    float r = red[0]; __syncthreads();
    return r;
}

// ---------------------------------------------------------------------------
// setup: write_idx per row, valid_max, new_position output
// ---------------------------------------------------------------------------
__global__ void setup_kernel(const int* __restrict__ pos, int* __restrict__ meta,
                             float* __restrict__ out_pos) {
    int t = threadIdx.x;
    if (t < BATCH) {
        meta[t] = pos[t] & (MAXP - 1);               // pos % MAX
        out_pos[t] = (float)pos[t] + 1.0f;           // new_position
    }
    if (t == 0) {
        int vm = 0;
        for (int b = 0; b < BATCH; ++b) {
            int v = pos[b] + 1; if (v > MAXP) v = MAXP;
            if (v > vm) vm = v;
        }
        meta[8] = vm;                                 // valid_max (shared mask)
    }
}

// ---------------------------------------------------------------------------
// bulk self-cache copy -> out (268 MB, pure streaming, float4)
// ---------------------------------------------------------------------------
__global__ void copy_cache_kernel(const float4* __restrict__ src,
                                  float4* __restrict__ dst, long n4) {
    long i = (long)blockIdx.x * blockDim.x + threadIdx.x;
    long stride = (long)gridDim.x * blockDim.x;
    for (; i < n4; i += stride) dst[i] = src[i];
}

// ---------------------------------------------------------------------------
// embedding + LN -> x (rows 0..7), zero pad rows 8..15
// ---------------------------------------------------------------------------
__global__ __launch_bounds__(256)
void embed_kernel(const int* __restrict__ ids, const int* __restrict__ pos,
                  const float* __restrict__ tok, const float* __restrict__ pe,
                  const float* __restrict__ scale, const float* __restrict__ bias,
                  float* __restrict__ x) {
    int row = blockIdx.x;
    int t = threadIdx.x;
    if (row >= BATCH) {  // zero pad rows for WMMA M=16
        for (int i = t; i < EMB; i += 256) x[row * EMB + i] = 0.f;
        return;
    }
    __shared__ float red[256];
    const float* tr = tok + (long)ids[row] * EMB;
    const float* pr = pe + (long)pos[row] * EMB;
    float e[4]; float sum = 0.f, sq = 0.f;
    for (int i = 0; i < 4; ++i) {
        int j = t + i * 256;
        float v = tr[j] + pr[j];
        e[i] = v; sum += v; sq += v * v;
    }
    sum = block_reduce_sum256(sum, red);
    sq  = block_reduce_sum256(sq, red);
    float mean = sum * (1.f / EMB);
    float var  = sq * (1.f / EMB) - mean * mean;
    float rstd = rsqrtf(var + 1e-5f);
    for (int i = 0; i < 4; ++i) {
        int j = t + i * 256;
        x[row * EMB + j] = (e[i] - mean) * rstd * scale[j] + bias[j];
    }
}

// ---------------------------------------------------------------------------
// residual + LayerNorm: out = LN(a + badd) * scale + bias   (rows 0..7)
// ---------------------------------------------------------------------------
__global__ __launch_bounds__(256)
void ln_residual_kernel(const float* __restrict__ a, const float* __restrict__ badd,
                        const float* __restrict__ scale, const float* __restrict__ bias,
                        float* __restrict__ out) {
    __shared__ float red[256];
    int row = blockIdx.x, t = threadIdx.x;
    float e[4]; float sum = 0.f, sq = 0.f;
    for (int i = 0; i < 4; ++i) {
        int j = t + i * 256;
        float v = a[row * EMB + j];
        if (badd) v += badd[row * EMB + j];
        e[i] = v; sum += v; sq += v * v;
    }
    sum = block_reduce_sum256(sum, red);
    sq  = block_reduce_sum256(sq, red);
    float mean = sum * (1.f / EMB);
    float var  = sq * (1.f / EMB) - mean * mean;
    float rstd = rsqrtf(var + 1e-5f);
    for (int i = 0; i < 4; ++i) {
        int j = t + i * 256;
        out[row * EMB + j] = (e[i] - mean) * rstd * scale[j] + bias[j];
    }
}

// ---------------------------------------------------------------------------
// Y[16xN] = X[16xK] @ W[KxN] (+bias, optional relu) via V_WMMA_F32_16X16X4_F32
// 8 waves/block, one 16-col tile per wave.
// A 16x4 f32 layout: lane m=lane&15; VGPR0/1 hold K = 2*(lane>>4) + {0,1}
// B 4x16  layout:    lane n=lane&15; VGPR0/1 hold K = 2*(lane>>4) + {0,1}
// C 16x16 f32: VGPR j -> row (j + 8*(lane>>4)), col lane&15
// ---------------------------------------------------------------------------
__global__ __launch_bounds__(256)
void gemm16_wmma(const float* __restrict__ X, const float* __restrict__ W,
                 const float* __restrict__ bias, float* __restrict__ Y,
                 int K, int N, int relu) {
    int wave = threadIdx.x >> 5;
    int lane = threadIdx.x & 31;
    int n0 = (blockIdx.x * 8 + wave) * 16;
    int m  = lane & 15;
    int kh = (lane >> 4) << 1;
    int n  = lane & 15;
    v8f c = {0.f, 0.f, 0.f, 0.f, 0.f, 0.f, 0.f, 0.f};
    const float* xp = X + m * K + kh;
    const float* wp = W + (long)kh * N + n0 + n;
    for (int k0 = 0; k0 < K; k0 += 4) {
        v2f a = *(const v2f*)xp;                 // A[m][k], A[m][k+1]
        v2f b; b.x = wp[0]; b.y = wp[N];         // B[k][n], B[k+1][n]
        c = __builtin_amdgcn_wmma_f32_16x16x4_f32(
                false, a, false, b, (short)0, c, false, false);
        xp += 4;
        wp += (long)4 * N;
    }
    int half = lane >> 4;
    float bv = bias ? bias[n0 + n] : 0.f;
    for (int j = 0; j < 8; ++j) {
        int row = j + half * 8;
        float v = c[j] + bv;
        if (relu) v = fmaxf(v, 0.f);
        Y[(long)row * N + n0 + n] = v;
    }
}

// ---------------------------------------------------------------------------
// scatter fresh k/v rows into the output cache slot [l, c, b, widx[b], :, :]
// ---------------------------------------------------------------------------
__global__ void scatter_kv_kernel(const float* __restrict__ kws,
                                  const float* __restrict__ vws,
                                  const int* __restrict__ meta,
                                  float* __restrict__ cache_out, int l) {
    int idx = blockIdx.x * 256 + threadIdx.x;     // 0 .. 16383
    int c = idx >> 13;                            // 0 = K, 1 = V
    int r = idx & 8191;
    int b = r >> 10;
    int j = r & 1023;                             // h*128 + d
    const float* src = c ? vws : kws;
    long dst = (((long)(l * 2 + c) * BATCH + b) * MAXP + meta[b]) * (HNUM * DHEAD) + j;
    cache_out[dst] = src[b * EMB + j];
}

// ---------------------------------------------------------------------------
// attention: one block per (b,h). SELF: patch key row widx with fresh k/v and
// apply kmask (key < valid_max). CROSS: dense over NK keys.
// ---------------------------------------------------------------------------
template <int NK, bool SELF>
__global__ __launch_bounds__(256)
void attn_kernel(const float* __restrict__ qws, const float* __restrict__ kws,
                 const float* __restrict__ vws, const float* __restrict__ kc,
                 const float* __restrict__ vc, const int* __restrict__ meta,
                 float* __restrict__ outws) {
    __shared__ float qsh[DHEAD];
    __shared__ float ssh[NK];
    __shared__ float red[256];
    int b = blockIdx.x / HNUM, h = blockIdx.x % HNUM;
    int t = threadIdx.x;
    int widx = SELF ? meta[b] : -1;
    int vmax = SELF ? meta[8] : NK;
    const float scale = 0.08838834764831845f;     // 1/sqrt(128)

    if (t < DHEAD) qsh[t] = qws[b * EMB + h * DHEAD + t];
    __syncthreads();

    // scores
    for (int key = t; key < NK; key += 256) {
        const float* kr;
        if (SELF && key == widx) kr = kws + b * EMB + h * DHEAD;
        else kr = kc + (((long)b * NK + key) * HNUM + h) * DHEAD;
        const float4* kr4 = (const float4*)kr;
        float acc = 0.f;
        #pragma unroll 8
        for (int d4 = 0; d4 < DHEAD / 4; ++d4) {
            float4 kv = kr4[d4];
            acc += qsh[4 * d4 + 0] * kv.x + qsh[4 * d4 + 1] * kv.y +
                   qsh[4 * d4 + 2] * kv.z + qsh[4 * d4 + 3] * kv.w;
        }
        ssh[key] = (key < vmax) ? acc * scale : -1e9f;
    }
    __syncthreads();

    // softmax: max
    float mx = -3.4e38f;
    for (int key = t; key < NK; key += 256) mx = fmaxf(mx, ssh[key]);
    red[t] = mx; __syncthreads();
    for (int s = 128; s > 0; s >>= 1) {
        if (t < s) red[t] = fmaxf(red[t], red[t + s]);
        __syncthreads();
    }
    mx = red[0]; __syncthreads();

    // exp + sum
    float psum = 0.f;
    for (int key = t; key < NK; key += 256) {
        float p = expf(ssh[key] - mx);
        ssh[key] = p; psum += p;
    }
    psum = block_reduce_sum256(psum, red);
    float inv = 1.f / psum;

    // weighted value sum, split keys across two half-blocks
    int d = t & (DHEAD - 1);
    int grp = t >> 7;
    float acc = 0.f;
    for (int key = grp * (NK / 2); key < (grp + 1) * (NK / 2); ++key) {
        const float* vr;
        if (SELF && key == widx) vr = vws + b * EMB + h * DHEAD;
        else vr = vc + (((long)b * NK + key) * HNUM + h) * DHEAD;
        acc += ssh[key] * vr[d];
    }
    red[t] = acc; __syncthreads();
    if (t < DHEAD)
        outws[b * EMB + h * DHEAD + t] = (red[t] + red[t + 128]) * inv;
}

// ---------------------------------------------------------------------------
extern "C" void kernel_launch(void* const* d_in, const int* in_sizes, int n_in,
                              void* d_out, int out_size, void* d_ws, size_t ws_size,
                              hipStream_t stream) {
    (void)in_sizes; (void)n_in; (void)out_size; (void)ws_size;
    const int*   ids    = (const int*)d_in[0];
    const int*   pos    = (const int*)d_in[1];
    // d_in[2] encoder_hidden_states: dead (projection result unused) -> skipped
    const float* selfc  = (const float*)d_in[3];
    const float* crossc = (const float*)d_in[4];
    const float* tok    = (const float*)d_in[5];
    const float* pe     = (const float*)d_in[6];
    const float* embs   = (const float*)d_in[7];
    const float* embb   = (const float*)d_in[8];
    // d_in[9], d_in[10]: enc_proj_w/b dead
    const float* qkvw   = (const float*)d_in[11];
    const float* qkvb   = (const float*)d_in[12];
    const float* soutw  = (const float*)d_in[13];
    const float* soutb  = (const float*)d_in[14];
    const float* cqw    = (const float*)d_in[15];
    const float* cqb    = (const float*)d_in[16];
    const float* coutw  = (const float*)d_in[17];
    const float* coutb  = (const float*)d_in[18];
    const float* lns    = (const float*)d_in[19];
    const float* lnb    = (const float*)d_in[20];
    const float* w1     = (const float*)d_in[21];
    const float* b1     = (const float*)d_in[22];
    const float* w2     = (const float*)d_in[23];
    const float* b2     = (const float*)d_in[24];
    const float* fls    = (const float*)d_in[25];
    const float* flb    = (const float*)d_in[26];

    float* ws   = (float*)d_ws;
    float* x    = ws + WS_X;
    float* y    = ws + WS_Y;
    float* q    = ws + WS_Q;
    float* k    = ws + WS_K;
    float* v    = ws + WS_V;
    float* attn = ws + WS_ATTN;
    float* h1   = ws + WS_H1;
    int*   meta = (int*)(ws + WS_META);

    float* out_hidden = (float*)d_out;                  // (B,1,E)
    float* out_pos    = out_hidden + BATCH * EMB;       // (B,)
    float* out_cache  = out_pos + BATCH;                // (L,2,B,MAX,H,D)

    // 1) bulk cache copy (268 MB each way; slots patched later per layer)
    long n4 = (long)LNUM * 2 * BATCH * MAXP * HNUM * DHEAD / 4;
    copy_cache_kernel<<<8192, 256, 0, stream>>>((const float4*)selfc,
                                                (float4*)out_cache, n4);
    // 2) metadata + new_position
    setup_kernel<<<1, 256, 0, stream>>>(pos, meta, out_pos);
    // 3) embedding + LN -> x (pad rows zeroed)
    embed_kernel<<<16, 256, 0, stream>>>(ids, pos, tok, pe, embs, embb, x);

    for (int l = 0; l < LNUM; ++l) {
        // fused QKV projection (three 1024x1024 WMMA GEMMs)
        gemm16_wmma<<<8, 256, 0, stream>>>(x, qkvw + (long)(l * 3 + 0) * EMB * EMB,
                                           qkvb + (l * 3 + 0) * EMB, q, EMB, EMB, 0);
        gemm16_wmma<<<8, 256, 0, stream>>>(x, qkvw + (long)(l * 3 + 1) * EMB * EMB,
                                           qkvb + (l * 3 + 1) * EMB, k, EMB, EMB, 0);
        gemm16_wmma<<<8, 256, 0, stream>>>(x, qkvw + (long)(l * 3 + 2) * EMB * EMB,
                                           qkvb + (l * 3 + 2) * EMB, v, EMB, EMB, 0);
        // write new k/v into output cache ring slot
        scatter_kv_kernel<<<64, 256, 0, stream>>>(k, v, meta, out_cache, l);
        // self attention over 512 cached keys (read original cache, patch widx)
        const float* kcl = selfc + (long)(l * 2 + 0) * BATCH * MAXP * HNUM * DHEAD;
        const float* vcl = selfc + (long)(l * 2 + 1) * BATCH * MAXP * HNUM * DHEAD;
        attn_kernel<MAXP, true><<<BATCH * HNUM, 256, 0, stream>>>(q, k, v, kcl, vcl,
                                                                  meta, attn);
        gemm16_wmma<<<8, 256, 0, stream>>>(attn, soutw + (long)l * EMB * EMB,
                                           soutb + l * EMB, y, EMB, EMB, 0);
        ln_residual_kernel<<<8, 256, 0, stream>>>(x, y, lns + (l * 3 + 0) * EMB,
                                                  lnb + (l * 3 + 0) * EMB, x);
        // cross attention over 1024 encoder keys
        gemm16_wmma<<<8, 256, 0, stream>>>(x, cqw + (long)l * EMB * EMB,
                                           cqb + l * EMB, q, EMB, EMB, 0);
        const float* ckc = crossc + (long)(l * 2 + 0) * BATCH * ENCL * HNUM * DHEAD;
        const float* cvc = crossc + (long)(l * 2 + 1) * BATCH * ENCL * HNUM * DHEAD;
        attn_kernel<ENCL, false><<<BATCH * HNUM, 256, 0, stream>>>(q, k, v, ckc, cvc,
                                                                   meta, attn);
        gemm16_wmma<<<8, 256, 0, stream>>>(attn, coutw + (long)l * EMB * EMB,
                                           coutb + l * EMB, y, EMB, EMB, 0);
        ln_residual_kernel<<<8, 256, 0, stream>>>(x, y, lns + (l * 3 + 1) * EMB,
                                                  lnb + (l * 3 + 1) * EMB, x);
        // FFN: relu(x@W1+b1)@W2+b2
        gemm16_wmma<<<32, 256, 0, stream>>>(x, w1 + (long)l * EMB * DFFN,
                                            b1 + l * DFFN, h1, EMB, DFFN, 1);
        gemm16_wmma<<<8, 256, 0, stream>>>(h1, w2 + (long)l * DFFN * EMB,
                                           b2 + l * EMB, y, DFFN, EMB, 0);
        ln_residual_kernel<<<8, 256, 0, stream>>>(x, y, lns + (l * 3 + 2) * EMB,
                                                  lnb + (l * 3 + 2) * EMB, x);
    }
    // final LN -> hidden output
    ln_residual_kernel<<<8, 256, 0, stream>>>(x, nullptr, fls, flb, out_hidden);
}